// GCN_5652176962021
// MI455X (gfx1250) — compile-verified
//
#include <hip/hip_runtime.h>
#include <hip/hip_bf16.h>

typedef float v2f __attribute__((ext_vector_type(2)));
typedef float v8f __attribute__((ext_vector_type(8)));

#define N_FEAT   128
#define HIDDEN   64
#define HID4     (HIDDEN / 4)   // float4 chunks per row
#define N_GRAPHS 256

// ---------------------------------------------------------------------------
// Utility kernels
// ---------------------------------------------------------------------------
__global__ void fill_kernel(float* __restrict__ p, float v, long long n) {
  long long i = (long long)blockIdx.x * blockDim.x + threadIdx.x;
  if (i < n) p[i] = v;
}

__global__ void deg_kernel(const int* __restrict__ dst, float* __restrict__ deg, int n_edges) {
  int e = blockIdx.x * blockDim.x + threadIdx.x;
  if (e < n_edges) atomicAdd(&deg[dst[e]], 1.0f);
}

__global__ void rsqrt_kernel(float* __restrict__ p, int n) {
  int i = blockIdx.x * blockDim.x + threadIdx.x;
  if (i < n) {
    float d = p[i];
    p[i] = (d > 0.0f) ? rsqrtf(d) : 0.0f;
  }
}

// ---------------------------------------------------------------------------
// WMMA f32 GEMM: H[n_rows, 64] = X[n_rows, K] @ W[K, 64]
// One wave computes a 16x16 output tile using V_WMMA_F32_16X16X4_F32.
// A fragment (16x4 f32): lanes 0-15 -> M=lane, {v0=K0, v1=K1};
//                        lanes 16-31 -> M=lane-16, {v0=K2, v1=K3}.
// B fragment mirrored (4x16, N=lane within half).
// C/D: VGPR r: lanes 0-15 -> M=r, lanes 16-31 -> M=r+8, N=lane%16.
// ---------------------------------------------------------------------------
template <int K>
__global__ __launch_bounds__(128)
void gemm_wmma_f32(const float* __restrict__ X, const float* __restrict__ W,
                   float* __restrict__ H, int n_rows) {
  const int wave = threadIdx.x >> 5;      // 0..3  -> N tile (HIDDEN/16 = 4)
  const int lane = threadIdx.x & 31;
  const int half = lane >> 4;             // 0 or 1
  const int lm   = lane & 15;
  const int m0   = blockIdx.x * 16;
  const int n0   = wave * 16;
  if (m0 >= n_rows) return;

  v8f acc = {};
  const float* arow = X + (size_t)(m0 + lm) * K + 2 * half;
  const float* bcol = W + (size_t)(2 * half) * HIDDEN + n0 + lm;

#pragma unroll 8
  for (int k = 0; k < K; k += 4) {
    v2f a, b;
    a.x = arow[k];
    a.y = arow[k + 1];
    b.x = bcol[(size_t)k * HIDDEN];
    b.y = bcol[(size_t)(k + 1) * HIDDEN];
    acc = __builtin_amdgcn_wmma_f32_16x16x4_f32(
        /*neg_a=*/false, a, /*neg_b=*/false, b,
        /*c_mod=*/(short)0, acc, /*reuse_a=*/false, /*reuse_b=*/false);
  }

  float* hrow = H + (size_t)(m0 + half * 8) * HIDDEN + n0 + lm;
#pragma unroll
  for (int r = 0; r < 8; ++r) hrow[(size_t)r * HIDDEN] = acc[r];
}

// ---------------------------------------------------------------------------
// Self-loop + bias init:  out[n, q*4..q*4+3] = h[n,...]*dinv[n]^2 + b[...]
// One thread per (node, float4 chunk).
// ---------------------------------------------------------------------------
__global__ void node_init_kernel(const float4* __restrict__ h4, const float* __restrict__ dinv,
                                 const float4* __restrict__ b4, float4* __restrict__ out4,
                                 int n_nodes) {
  long long t = (long long)blockIdx.x * blockDim.x + threadIdx.x;
  int n = (int)(t >> 4), q = (int)(t & 15);
  if (n >= n_nodes) return;
  float di = dinv[n];
  float s = di * di;
  float4 v = h4[(size_t)n * HID4 + q];
  float4 b = b4[q];
  float4 r;
  r.x = v.x * s + b.x;
  r.y = v.y * s + b.y;
  r.z = v.z * s + b.z;
  r.w = v.w * s + b.w;
  out4[(size_t)n * HID4 + q] = r;
}

// ---------------------------------------------------------------------------
// Edge scatter: out[dst, :] += h[src, :] * dinv[src]*dinv[dst]
// 16 threads per edge; each gathers one float4 (global_load_b128) and issues
// 4 global_atomic_add_f32 into the L2 atomic units.
// ---------------------------------------------------------------------------
__global__ void edge_scatter_kernel(const float4* __restrict__ h4, const float* __restrict__ dinv,
                                    const int* __restrict__ src, const int* __restrict__ dst,
                                    float* __restrict__ out, int n_edges) {
  long long t = (long long)blockIdx.x * blockDim.x + threadIdx.x;
  int e = (int)(t >> 4), q = (int)(t & 15);
  if (e >= n_edges) return;
  int s = src[e], d = dst[e];
  float norm = dinv[s] * dinv[d];
  float4 v = h4[(size_t)s * HID4 + q];
  float* o = out + (size_t)d * HIDDEN + q * 4;
  atomicAdd(o + 0, v.x * norm);
  atomicAdd(o + 1, v.y * norm);
  atomicAdd(o + 2, v.z * norm);
  atomicAdd(o + 3, v.w * norm);
}

__global__ void relu_kernel(float4* __restrict__ p, long long n4) {
  long long i = (long long)blockIdx.x * blockDim.x + threadIdx.x;
  if (i < n4) {
    float4 v = p[i];
    v.x = fmaxf(v.x, 0.0f);
    v.y = fmaxf(v.y, 0.0f);
    v.z = fmaxf(v.z, 0.0f);
    v.w = fmaxf(v.w, 0.0f);
    p[i] = v;
  }
}

// ---------------------------------------------------------------------------
// Pooling: per (node, float4 chunk) atomics into per-graph sums; q==0 counts.
// ---------------------------------------------------------------------------
__global__ void pool_kernel(const float4* __restrict__ h4, const int* __restrict__ batch,
                            float* __restrict__ sums, float* __restrict__ cnt, int n_nodes) {
  long long t = (long long)blockIdx.x * blockDim.x + threadIdx.x;
  int n = (int)(t >> 4), q = (int)(t & 15);
  if (n >= n_nodes) return;
  int g = batch[n];
  float4 v = h4[(size_t)n * HID4 + q];
  float* sg = sums + (size_t)g * HIDDEN + q * 4;
  atomicAdd(sg + 0, v.x);
  atomicAdd(sg + 1, v.y);
  atomicAdd(sg + 2, v.z);
  atomicAdd(sg + 3, v.w);
  if (q == 0) atomicAdd(&cnt[g], 1.0f);
}

// Final head: out[g] = dot(sums[g]/max(cnt,1), Wl) + bl   (256 graphs, 1 block)
__global__ void head_kernel(const float* __restrict__ sums, const float* __restrict__ cnt,
                            const float* __restrict__ Wl, const float* __restrict__ bl,
                            float* __restrict__ out) {
  int g = threadIdx.x;
  if (g >= N_GRAPHS) return;
  float c = fmaxf(cnt[g], 1.0f);
  float acc = 0.0f;
#pragma unroll 8
  for (int f = 0; f < HIDDEN; ++f) acc += (sums[(size_t)g * HIDDEN + f] / c) * Wl[f];
  out[g] = acc + bl[0];
}

// ---------------------------------------------------------------------------
static inline unsigned grid1(long long total, int block) {
  return (unsigned)((total + block - 1) / block);
}

extern "C" void kernel_launch(void* const* d_in, const int* in_sizes, int n_in,
                              void* d_out, int out_size, void* d_ws, size_t ws_size,
                              hipStream_t stream) {
  const float* x      = (const float*)d_in[0];
  const int*   eidx   = (const int*)d_in[1];
  const int*   batch  = (const int*)d_in[2];
  const float* W1     = (const float*)d_in[3];
  const float* b1     = (const float*)d_in[4];
  const float* W2     = (const float*)d_in[5];
  const float* b2     = (const float*)d_in[6];
  const float* Wl     = (const float*)d_in[7];
  const float* bl     = (const float*)d_in[8];
  float*       out    = (float*)d_out;

  const int n_nodes = in_sizes[0] / N_FEAT;
  const int n_edges = in_sizes[1] / 2;
  const int* src = eidx;
  const int* dst = eidx + n_edges;

  // Workspace layout (floats; all segments stay 16-byte aligned)
  float* dinv = (float*)d_ws;                         // n_nodes
  float* bufA = dinv + n_nodes;                       // n_nodes*64  (GEMM output h)
  float* bufB = bufA + (size_t)n_nodes * HIDDEN;      // n_nodes*64  (scatter accumulator)
  float* sums = bufB + (size_t)n_nodes * HIDDEN;      // 256*64
  float* cnt  = sums + (size_t)N_GRAPHS * HIDDEN;     // 256

  const long long n_nq = (long long)n_nodes * HID4;   // node*float4-chunk work items
  const long long n_eq = (long long)n_edges * HID4;   // edge*float4-chunk work items
  const unsigned gemm_grid = (unsigned)((n_nodes + 15) / 16);

  // ---- degree -> dinv -------------------------------------------------
  fill_kernel<<<grid1(n_nodes, 256), 256, 0, stream>>>(dinv, 1.0f, n_nodes); // self loops
  deg_kernel<<<grid1(n_edges, 256), 256, 0, stream>>>(dst, dinv, n_edges);
  rsqrt_kernel<<<grid1(n_nodes, 256), 256, 0, stream>>>(dinv, n_nodes);

  // ---- layer 1: h = relu(Dinv (A+I) Dinv (x@W1) + b1) -----------------
  gemm_wmma_f32<N_FEAT><<<gemm_grid, 128, 0, stream>>>(x, W1, bufA, n_nodes);
  node_init_kernel<<<grid1(n_nq, 256), 256, 0, stream>>>(
      (const float4*)bufA, dinv, (const float4*)b1, (float4*)bufB, n_nodes);
  edge_scatter_kernel<<<grid1(n_eq, 256), 256, 0, stream>>>(
      (const float4*)bufA, dinv, src, dst, bufB, n_edges);
  relu_kernel<<<grid1(n_nq, 256), 256, 0, stream>>>((float4*)bufB, n_nq);

  // ---- layer 2: h = Dinv (A+I) Dinv (h@W2) + b2 -----------------------
  gemm_wmma_f32<HIDDEN><<<gemm_grid, 128, 0, stream>>>(bufB, W2, bufA, n_nodes);
  node_init_kernel<<<grid1(n_nq, 256), 256, 0, stream>>>(
      (const float4*)bufA, dinv, (const float4*)b2, (float4*)bufB, n_nodes);
  edge_scatter_kernel<<<grid1(n_eq, 256), 256, 0, stream>>>(
      (const float4*)bufA, dinv, src, dst, bufB, n_edges);

  // ---- global mean pool + linear head ---------------------------------
  fill_kernel<<<grid1(N_GRAPHS * HIDDEN + N_GRAPHS, 256), 256, 0, stream>>>(
      sums, 0.0f, N_GRAPHS * HIDDEN + N_GRAPHS);
  pool_kernel<<<grid1(n_nq, 256), 256, 0, stream>>>(
      (const float4*)bufB, batch, sums, cnt, n_nodes);
  head_kernel<<<1, N_GRAPHS, 0, stream>>>(sums, cnt, Wl, bl, out);
}